// GNN_21105469292716
// MI455X (gfx1250) — compile-verified
//
#include <hip/hip_runtime.h>

// ---------------------------------------------------------------------------
// GraphSAGE 2-layer forward for MI455X (gfx1250, wave32).
// Aggregation is linear => project-then-aggregate: y = x@Wl first (WMMA),
// then scatter 16 (resp 2) floats per edge instead of 64. Edge scatter uses
// global_atomic_add_f32; deg shared between both layers.
// ---------------------------------------------------------------------------

typedef __attribute__((ext_vector_type(2))) float v2f;
typedef __attribute__((ext_vector_type(4))) float v4f;
typedef __attribute__((ext_vector_type(8))) float v8f;

static __device__ __forceinline__ v8f wmma_f32x4(v2f a, v2f b, v8f c) {
  // D = A(16x4 f32) * B(4x16 f32) + C(16x16 f32)
  return __builtin_amdgcn_wmma_f32_16x16x4_f32(
      /*neg_a=*/false, a, /*neg_b=*/false, b,
      /*c_mod=*/(short)0, c, /*reuse_a=*/false, /*reuse_b=*/false);
}

// ------------------------------- zero ---------------------------------------
__global__ void k_zero(float* __restrict__ p, long long nElem) {
  long long i = (long long)blockIdx.x * blockDim.x + threadIdx.x;
  if (i < nElem) p[i] = 0.0f;
}

// --------------------- GEMM1: [n,64] x [64,16] (x2 weights) -----------------
#define T1_STRIDE 68  // 64 + pad: rows land on distinct LDS banks, keeps 16B align
__launch_bounds__(256)
__global__ void k_gemm1(const float* __restrict__ x, const float* __restrict__ Wl,
                        const float* __restrict__ Wr, float* __restrict__ y1,
                        float* __restrict__ z1, int n) {
  __shared__ float tile[128 * T1_STRIDE];
  const int tid = threadIdx.x;
  const int rb  = blockIdx.x * 128;

  // coalesced global b128 loads -> LDS (128 rows x 64 cols)
#pragma unroll
  for (int it = 0; it < 8; ++it) {
    int idx = (it * 256 + tid) * 4;
    int r = idx >> 6, c = idx & 63;
    int grow = rb + r; if (grow > n - 1) grow = n - 1;
    v4f v = *(const v4f*)(x + (size_t)grow * 64 + c);
    *(v4f*)(tile + r * T1_STRIDE + c) = v;
  }
  __syncthreads();

  const int lane = tid & 31;
  const int wave = tid >> 5;
  const int mr   = lane & 15;           // M (A) / N (B,C) index
  const int kh   = (lane >> 4) << 1;    // K half: 0 or 2

  // B fragments for both weight matrices: b[s][j] = W[(4s+kh+j)*16 + mr]
  v2f bl[16], br[16];
#pragma unroll
  for (int s = 0; s < 16; ++s) {
    int k0 = 4 * s + kh;
    v2f tl, tr;
    tl.x = Wl[k0 * 16 + mr]; tl.y = Wl[(k0 + 1) * 16 + mr];
    tr.x = Wr[k0 * 16 + mr]; tr.y = Wr[(k0 + 1) * 16 + mr];
    bl[s] = tl; br[s] = tr;
  }

  const float* arow = tile + (wave * 16 + mr) * T1_STRIDE;
  v8f cl = {}; v8f cr = {};
#pragma unroll
  for (int s = 0; s < 16; ++s) {
    v2f a = *(const v2f*)(arow + 4 * s + kh);
    cl = wmma_f32x4(a, bl[s], cl);
    cr = wmma_f32x4(a, br[s], cr);
  }

  // C layout: VGPR v, lanes 0-15 -> M=v ; lanes 16-31 -> M=v+8 ; N=mr
  const int rowHalf = (lane >> 4) << 3;
#pragma unroll
  for (int v = 0; v < 8; ++v) {
    int row = rb + wave * 16 + rowHalf + v;
    if (row < n) {
      y1[(size_t)row * 16 + mr] = cl[v];
      z1[(size_t)row * 16 + mr] = cr[v];
    }
  }
}

// ------------------- edge scatter, layer 1 (16 feats/edge) ------------------
__global__ void k_scatter1(const int* __restrict__ ei, const float* __restrict__ y1,
                           float* __restrict__ agg1, float* __restrict__ deg,
                           long long E) {
  long long t = (long long)blockIdx.x * blockDim.x + threadIdx.x;
  if (t >= E * 16) return;
  long long e = t >> 4;
  int f = (int)(t & 15);
  int src = ei[e];
  int dst = ei[E + e];
  float v = y1[(size_t)src * 16 + f];
  atomicAdd(&agg1[(size_t)dst * 16 + f], v);
  if (f == 0) atomicAdd(&deg[dst], 1.0f);
}

// -------------------- h = relu(agg1/max(deg,1) + z1 + b1) -------------------
__global__ void k_combine1(const float* __restrict__ agg1, const float* __restrict__ deg,
                           const float* __restrict__ z1, const float* __restrict__ b1,
                           float* __restrict__ h, int n) {
  long long i = (long long)blockIdx.x * blockDim.x + threadIdx.x;
  if (i >= (long long)n * 16) return;
  int node = (int)(i >> 4);
  int f = (int)(i & 15);
  float d = fmaxf(deg[node], 1.0f);
  float v = agg1[i] / d + z1[i] + b1[f];
  h[i] = fmaxf(v, 0.0f);
}

// --------------------- GEMM2: [n,16] x [16,2] (x2 weights) ------------------
#define T2_STRIDE 20  // 16 + pad, keeps 16B alignment and spreads banks
__launch_bounds__(256)
__global__ void k_gemm2(const float* __restrict__ h, const float* __restrict__ Wl,
                        const float* __restrict__ Wr, float* __restrict__ y2,
                        float* __restrict__ z2, int n) {
  __shared__ float tile[128 * T2_STRIDE];
  const int tid = threadIdx.x;
  const int rb  = blockIdx.x * 128;

#pragma unroll
  for (int it = 0; it < 2; ++it) {
    int idx = (it * 256 + tid) * 4;
    int r = idx >> 4, c = idx & 15;
    int grow = rb + r; if (grow > n - 1) grow = n - 1;
    v4f v = *(const v4f*)(h + (size_t)grow * 16 + c);
    *(v4f*)(tile + r * T2_STRIDE + c) = v;
  }
  __syncthreads();

  const int lane = tid & 31;
  const int wave = tid >> 5;
  const int mr   = lane & 15;
  const int kh   = (lane >> 4) << 1;

  // B padded 16x2 -> 16x16 (cols >= 2 are zero)
  v2f bl[4], br[4];
#pragma unroll
  for (int s = 0; s < 4; ++s) {
    int k0 = 4 * s + kh;
    v2f tl = {}; v2f tr = {};
    if (mr < 2) {
      tl.x = Wl[k0 * 2 + mr]; tl.y = Wl[(k0 + 1) * 2 + mr];
      tr.x = Wr[k0 * 2 + mr]; tr.y = Wr[(k0 + 1) * 2 + mr];
    }
    bl[s] = tl; br[s] = tr;
  }

  const float* arow = tile + (wave * 16 + mr) * T2_STRIDE;
  v8f cl = {}; v8f cr = {};
#pragma unroll
  for (int s = 0; s < 4; ++s) {
    v2f a = *(const v2f*)(arow + 4 * s + kh);
    cl = wmma_f32x4(a, bl[s], cl);
    cr = wmma_f32x4(a, br[s], cr);
  }

  const int rowHalf = (lane >> 4) << 3;
  if (mr < 2) {
#pragma unroll
    for (int v = 0; v < 8; ++v) {
      int row = rb + wave * 16 + rowHalf + v;
      if (row < n) {
        y2[(size_t)row * 2 + mr] = cl[v];
        z2[(size_t)row * 2 + mr] = cr[v];
      }
    }
  }
}

// -------------------- edge scatter, layer 2 (2 feats/edge) ------------------
__global__ void k_scatter2(const int* __restrict__ ei, const float* __restrict__ y2,
                           float* __restrict__ agg2, long long E) {
  long long t = (long long)blockIdx.x * blockDim.x + threadIdx.x;
  if (t >= E * 2) return;
  long long e = t >> 1;
  int f = (int)(t & 1);
  int src = ei[e];
  int dst = ei[E + e];
  atomicAdd(&agg2[(size_t)dst * 2 + f], y2[(size_t)src * 2 + f]);
}

// ------------------ out = agg2/max(deg,1) + z2 + b2 -------------------------
__global__ void k_combine2(const float* __restrict__ agg2, const float* __restrict__ deg,
                           const float* __restrict__ z2, const float* __restrict__ b2,
                           float* __restrict__ out, int n) {
  long long i = (long long)blockIdx.x * blockDim.x + threadIdx.x;
  if (i >= (long long)n * 2) return;
  int node = (int)(i >> 1);
  int f = (int)(i & 1);
  float d = fmaxf(deg[node], 1.0f);
  out[i] = agg2[i] / d + z2[i] + b2[f];
}

// ---------------------------------------------------------------------------
extern "C" void kernel_launch(void* const* d_in, const int* in_sizes, int n_in,
                              void* d_out, int out_size, void* d_ws, size_t ws_size,
                              hipStream_t stream) {
  const float* x   = (const float*)d_in[0];
  const int*   ei  = (const int*)d_in[1];   // edge_index [2,E]
  const float* W1l = (const float*)d_in[2];
  const float* W1r = (const float*)d_in[3];
  const float* b1  = (const float*)d_in[4];
  const float* W2l = (const float*)d_in[5];
  const float* W2r = (const float*)d_in[6];
  const float* b2  = (const float*)d_in[7];
  float* out = (float*)d_out;

  const int       n = in_sizes[0] / 64;       // 100000
  const long long E = in_sizes[1] / 2;        // 1600000

  // workspace layout (floats): zeroed region first
  float* ws   = (float*)d_ws;
  float* agg1 = ws;                            // 16n
  float* deg  = agg1 + (size_t)16 * n;         // n
  float* agg2 = deg + n;                       // 2n
  float* y1   = agg2 + (size_t)2 * n;          // 16n
  float* z1   = y1 + (size_t)16 * n;           // 16n
  float* y2   = z1 + (size_t)16 * n;           // 2n
  float* z2   = y2 + (size_t)2 * n;            // 2n
  float* h    = y1;                            // reuse y1 after scatter1

  const long long nzero = (long long)19 * n;
  k_zero<<<(unsigned)((nzero + 255) / 256), 256, 0, stream>>>(ws, nzero);

  const unsigned gGemm = (unsigned)((n + 127) / 128);
  k_gemm1<<<gGemm, 256, 0, stream>>>(x, W1l, W1r, y1, z1, n);

  const long long t1 = E * 16;
  k_scatter1<<<(unsigned)((t1 + 255) / 256), 256, 0, stream>>>(ei, y1, agg1, deg, E);

  const long long c1 = (long long)n * 16;
  k_combine1<<<(unsigned)((c1 + 255) / 256), 256, 0, stream>>>(agg1, deg, z1, b1, h, n);

  k_gemm2<<<gGemm, 256, 0, stream>>>(h, W2l, W2r, y2, z2, n);

  const long long t2 = E * 2;
  k_scatter2<<<(unsigned)((t2 + 255) / 256), 256, 0, stream>>>(ei, y2, agg2, E);

  const long long c2 = (long long)n * 2;
  k_combine2<<<(unsigned)((c2 + 255) / 256), 256, 0, stream>>>(agg2, deg, z2, b2, out, n);
}